// Group_60155311948111
// MI455X (gfx1250) — compile-verified
//
#include <hip/hip_runtime.h>
#include <hip/hip_bf16.h>
#include <float.h>

#define N_PTS      131072
#define NUM_GROUP  2048
#define GROUP_SIZE 64

// ---- FPS config: multi-block with device-scope arrive/release barrier ----
#define FPS_NB   32
#define FPS_TPB  512
#define FPS_TOT  (FPS_NB * FPS_TPB)   // 16384 threads
#define FPS_PPT  (N_PTS / FPS_TOT)    // 8 points per thread

// ---- kNN config ----
#define KNN_THREADS 256
#define QT    16                 // queries per block (WMMA M)
#define NWAVE 8                  // waves per block
#define CHUNK (N_PTS / NWAVE)    // 16384 candidates per wave
#define TILES (CHUNK / 16)       // 1024 WMMA tiles per wave

typedef float v2f __attribute__((ext_vector_type(2)));
typedef float v8f __attribute__((ext_vector_type(8)));

struct FpsSync {
    unsigned count;
    unsigned phase;
    int      far;
    int      pad;
    float    pbv[FPS_NB];
    int      pbi[FPS_NB];
};

__global__ void fps_init_kernel(FpsSync* sync) {
    sync->count = 0u;
    sync->phase = 0u;
}

// ---------------------------------------------------------------------------
// Kernel 1: farthest point sampling across 32 co-resident blocks.
// Each thread owns a strided slice of dists[]; per iteration each block
// reduces its slice, the last block to arrive performs the deterministic
// global argmax (block order 0..31, tie -> smaller index) and releases.
// ---------------------------------------------------------------------------
__global__ __launch_bounds__(FPS_TPB)
void fps_kernel(const float* __restrict__ pc, const int* __restrict__ start,
                float* __restrict__ dists, float* __restrict__ centers,
                float* __restrict__ out_centers, FpsSync* sync) {
    __shared__ float s_rv[FPS_TPB / 32];
    __shared__ int   s_ri[FPS_TPB / 32];
    __shared__ float s_cx, s_cy, s_cz;
    __shared__ int   s_far;
    const int tid  = threadIdx.x;
    const int gtid = blockIdx.x * FPS_TPB + tid;

#pragma unroll
    for (int k = 0; k < FPS_PPT; ++k) dists[gtid + k * FPS_TOT] = FLT_MAX;
    if (tid == 0) s_far = start[0];
    __syncthreads();

    for (int it = 0; it < NUM_GROUP; ++it) {
        if (tid == 0) {
            const int f = s_far;
            const float cx = pc[3 * f], cy = pc[3 * f + 1], cz = pc[3 * f + 2];
            s_cx = cx; s_cy = cy; s_cz = cz;
            if (blockIdx.x == 0) {
                centers[3 * it] = cx; centers[3 * it + 1] = cy; centers[3 * it + 2] = cz;
                out_centers[3 * it] = cx; out_centers[3 * it + 1] = cy; out_centers[3 * it + 2] = cz;
            }
        }
        __syncthreads();
        const float cx = s_cx, cy = s_cy, cz = s_cz;

        float bv = -FLT_MAX;
        int   bi = 0;
#pragma unroll
        for (int k = 0; k < FPS_PPT; ++k) {
            const int i = gtid + k * FPS_TOT;        // ascending i -> first-max stable
            const float dx = pc[3 * i] - cx;
            const float dy = pc[3 * i + 1] - cy;
            const float dz = pc[3 * i + 2] - cz;
            const float d = dx * dx + dy * dy + dz * dz;
            const float old = dists[i];
            const float nd = d < old ? d : old;
            dists[i] = nd;
            if (nd > bv) { bv = nd; bi = i; }
        }
        for (int off = 16; off; off >>= 1) {
            const float ov = __shfl_down(bv, off, 32);
            const int   oi = __shfl_down(bi, off, 32);
            if (ov > bv || (ov == bv && oi < bi)) { bv = ov; bi = oi; }
        }
        const int wave = tid >> 5;
        if ((tid & 31) == 0) { s_rv[wave] = bv; s_ri[wave] = bi; }
        __syncthreads();

        if (tid == 0) {
            float v = s_rv[0]; int idx = s_ri[0];
            for (int w = 1; w < FPS_TPB / 32; ++w)
                if (s_rv[w] > v || (s_rv[w] == v && s_ri[w] < idx)) { v = s_rv[w]; idx = s_ri[w]; }
            sync->pbv[blockIdx.x] = v;
            sync->pbi[blockIdx.x] = idx;
            __threadfence();
            const unsigned old = atomicAdd(&sync->count, 1u);
            if (old == FPS_NB - 1) {           // sequencer: deterministic reduce
                __threadfence();
                float gv = sync->pbv[0]; int gi = sync->pbi[0];
                for (int b = 1; b < FPS_NB; ++b) {
                    const float v2 = sync->pbv[b]; const int i2 = sync->pbi[b];
                    if (v2 > gv || (v2 == gv && i2 < gi)) { gv = v2; gi = i2; }
                }
                sync->far   = gi;
                sync->count = 0u;
                __threadfence();
                __hip_atomic_store(&sync->phase, (unsigned)(it + 1),
                                   __ATOMIC_RELEASE, __HIP_MEMORY_SCOPE_AGENT);
                s_far = gi;
            } else {
                while (__hip_atomic_load(&sync->phase, __ATOMIC_ACQUIRE,
                                         __HIP_MEMORY_SCOPE_AGENT) <= (unsigned)it)
                    __builtin_amdgcn_s_sleep(2);
                s_far = sync->far;
            }
        }
        __syncthreads();
    }
}

// ---------------------------------------------------------------------------
// Kernel 2: kNN (k=64) + gather + center-subtract.
// 16 queries/block; each wave scans a 16384-candidate chunk in 16-wide tiles,
// one v_wmma_f32_16x16x4_f32 per tile (A=queries, B=-2*cands, C=q2+r2 => d2).
// Tails cached in registers; candidate loads software-pipelined (pc is
// L2-resident, 8 waves/block hide the remaining latency).
// ---------------------------------------------------------------------------
__device__ __forceinline__ unsigned f32key(float f) {
    const unsigned b = __float_as_uint(f);
    return (b & 0x80000000u) ? ~b : (b | 0x80000000u);   // monotone total order
}

__global__ __launch_bounds__(KNN_THREADS)
void knn_kernel(const float* __restrict__ pc, const float* __restrict__ centers,
                float* __restrict__ out_nb) {
    __shared__ float          s_dist[NWAVE][QT][GROUP_SIZE];
    __shared__ unsigned short s_rel[NWAVE][QT][GROUP_SIZE];
    __shared__ float          s_q[QT][4];
    __shared__ float          s_q2[QT];

    const int tid  = threadIdx.x;
    const int lane = tid & 31;
    const int wave = tid >> 5;
    const int q0   = blockIdx.x * QT;

    float* dflat = &s_dist[0][0][0];
    unsigned short* iflat = &s_rel[0][0][0];
    for (int i = tid; i < NWAVE * QT * GROUP_SIZE; i += KNN_THREADS) {
        dflat[i] = FLT_MAX;
        iflat[i] = 0;
    }
    if (tid < QT) {
        const float x = centers[(q0 + tid) * 3];
        const float y = centers[(q0 + tid) * 3 + 1];
        const float z = centers[(q0 + tid) * 3 + 2];
        s_q[tid][0] = x; s_q[tid][1] = y; s_q[tid][2] = z; s_q[tid][3] = 0.f;
        s_q2[tid] = x * x + y * y + z * z;
    }
    __syncthreads();

    const int  m  = lane & 15;
    const bool hi = (lane >= 16);

    // A (16x4 f32): lanes 0-15 hold K=0,1 (x,y); lanes 16-31 hold K=2,3 (z,0)
    v2f a;
    a.x = hi ? s_q[m][2] : s_q[m][0];
    a.y = hi ? 0.f       : s_q[m][1];

    float q2v[8], tailv[8];
#pragma unroll
    for (int r = 0; r < 8; ++r) {
        q2v[r]   = hi ? s_q2[r + 8] : s_q2[r];
        tailv[r] = FLT_MAX;
    }

    const int base = wave * CHUNK;
    // software pipeline: preload tile 0
    float x = pc[3 * (base + m)], y = pc[3 * (base + m) + 1], z = pc[3 * (base + m) + 2];

    for (int t = 0; t < TILES; ++t) {
        const float px = x, py = y, pz = z;
        if (t + 1 < TILES) {
            const int gn = base + (t + 1) * 16 + m;
            x = pc[3 * gn]; y = pc[3 * gn + 1]; z = pc[3 * gn + 2];
        }

        const float r2 = px * px + py * py + pz * pz;
        v2f b;                                     // B = -2 * candidate coords
        b.x = hi ? (-2.f * pz) : (-2.f * px);
        b.y = hi ? 0.f         : (-2.f * py);
        v8f c;
#pragma unroll
        for (int r = 0; r < 8; ++r) c[r] = q2v[r] + r2;

        v8f d = __builtin_amdgcn_wmma_f32_16x16x4_f32(
            false, a, false, b, (short)0, c, false, false);

        const int rel0 = t * 16;
#pragma unroll
        for (int r = 0; r < 8; ++r) {
            unsigned mask = (unsigned)__ballot(d[r] < tailv[r]);
            if (mask) {                                   // uniform, rare
                while (mask) {
                    const int L = __ffs(mask) - 1;
                    mask &= mask - 1;
                    const float val = __shfl(d[r], L, 32);
                    const int row = r + (L >= 16 ? 8 : 0);
                    const int rel = rel0 + (L & 15);
                    float*          dl = s_dist[wave][row];
                    unsigned short* il = s_rel[wave][row];
                    const float tail = dl[GROUP_SIZE - 1];
                    if (val < tail) {
                        const int e0 = lane, e1 = lane + 32;
                        const float d0 = dl[e0], d1 = dl[e1];
                        const unsigned short i0 = il[e0], i1 = il[e1];
                        const bool p0 = (d0 < val) || (d0 == val && (int)i0 < rel);
                        const bool p1 = (d1 < val) || (d1 == val && (int)i1 < rel);
                        const int pos = __popc((unsigned)__ballot(p0)) +
                                        __popc((unsigned)__ballot(p1));
                        __asm__ volatile("" ::: "memory");   // reads before shifts
                        if (e1 >= pos && e1 <= GROUP_SIZE - 2) { dl[e1 + 1] = d1; il[e1 + 1] = i1; }
                        if (e0 >= pos && e0 <= GROUP_SIZE - 2) { dl[e0 + 1] = d0; il[e0 + 1] = i0; }
                        __asm__ volatile("" ::: "memory");   // shifts before insert
                        if (lane == 0) { dl[pos] = val; il[pos] = (unsigned short)rel; }
                        __asm__ volatile("" ::: "memory");
                    }
                }
                tailv[r] = hi ? s_dist[wave][r + 8][GROUP_SIZE - 1]
                              : s_dist[wave][r][GROUP_SIZE - 1];
            }
        }
    }
    __syncthreads();

    // Merge: wave w exclusively owns queries 2w, 2w+1 (no block barriers).
    // Key = (ordered dist bits << 32) | entry position; entry position is
    // chunk-major so low bits already give global-index-stable tie-break.
    for (int qi = 0; qi < 2; ++qi) {
        const int q = wave * 2 + qi;
        const float cx = s_q[q][0], cy = s_q[q][1], cz = s_q[q][2];
        for (int sel = 0; sel < GROUP_SIZE; ++sel) {
            unsigned long long bestk = ~0ull;
#pragma unroll
            for (int j = 0; j < 16; ++j) {
                const int e = j * 32 + lane;       // 512 entries, bank-friendly
                const int w = e >> 6, jj = e & 63;
                const float dv = s_dist[w][q][jj];
                const unsigned long long key =
                    ((unsigned long long)f32key(dv) << 32) | (unsigned)e;
                if (key < bestk) bestk = key;
            }
            for (int off = 16; off; off >>= 1) {
                const unsigned long long ok = __shfl_down(bestk, off, 32);
                if (ok < bestk) bestk = ok;
            }
            if (lane == 0) {
                const int e = (int)(unsigned)(bestk & 0xFFFFFFFFull);
                const int w = e >> 6, jj = e & 63;
                const int gidx = w * CHUNK + (int)s_rel[w][q][jj];
                s_dist[w][q][jj] = FLT_MAX;        // consume entry
                const float vx = pc[3 * gidx], vy = pc[3 * gidx + 1], vz = pc[3 * gidx + 2];
                const int o = ((q0 + q) * GROUP_SIZE + sel) * 3;
                out_nb[o]     = vx - cx;
                out_nb[o + 1] = vy - cy;
                out_nb[o + 2] = vz - cz;
            }
            __asm__ volatile("" ::: "memory");     // invalidation visible next round
        }
    }
}

extern "C" void kernel_launch(void* const* d_in, const int* in_sizes, int n_in,
                              void* d_out, int out_size, void* d_ws, size_t ws_size,
                              hipStream_t stream) {
    const float* pc    = (const float*)d_in[0];
    const int*   start = (const int*)d_in[1];
    float* out = (float*)d_out;

    float* dists   = (float*)d_ws;            // N_PTS floats
    float* centers = dists + N_PTS;           // NUM_GROUP*3 floats
    FpsSync* sync  = (FpsSync*)(centers + NUM_GROUP * 3);

    float* out_nb  = out;                                 // [2048,64,3]
    float* out_ctr = out + NUM_GROUP * GROUP_SIZE * 3;    // [2048,3]

    fps_init_kernel<<<1, 1, 0, stream>>>(sync);
    fps_kernel<<<FPS_NB, FPS_TPB, 0, stream>>>(pc, start, dists, centers, out_ctr, sync);
    knn_kernel<<<NUM_GROUP / QT, KNN_THREADS, 0, stream>>>(pc, centers, out_nb);
}